// RIMDLoss_34703335752438
// MI455X (gfx1250) — compile-verified
//
#include <hip/hip_runtime.h>
#include <hip/hip_bf16.h>

typedef __attribute__((ext_vector_type(2))) float v2f;
typedef __attribute__((ext_vector_type(8))) float v8f;

#define TPB     256
#define NWAVES  (TPB / 32)
#define NBLK_E  2048
#define NBLK_N  512
#define NGRAPH  64

// ---------------------------------------------------------------------------
// Wave-level sum via V_WMMA_F32_16X16X4_F32.
// A = ones(16x4), B holds one value per lane (b[0]) and zero (b[1]).
// D[m][n] = sum_k B[k][n]  (column sums; all rows identical).
// D's VGPR0: lanes 0-15 -> row0 cols 0..15, lanes 16-31 -> row8 cols 0..15,
// so a 4-step xor-reduce within each 16-lane half yields the full 64-slot
// (i.e. 32-lane) total in every lane. Requires EXEC == all ones: all callers
// invoke this unconditionally from full 256-thread blocks.
// ---------------------------------------------------------------------------
__device__ __forceinline__ float wave_reduce_wmma(float v) {
    v2f a; a[0] = 1.0f; a[1] = 1.0f;
    v2f b; b[0] = v;    b[1] = 0.0f;
    v8f c = {};
    c = __builtin_amdgcn_wmma_f32_16x16x4_f32(false, a, false, b,
                                              (short)0, c, false, false);
    float s = c[0];
    s += __shfl_xor(s, 1, 32);
    s += __shfl_xor(s, 2, 32);
    s += __shfl_xor(s, 4, 32);
    s += __shfl_xor(s, 8, 32);
    return s;
}

// ---------------------------------------------------------------------------
// Edge kernel: stream edge_index, gather pred (L2-resident), accumulate
// S1 = sum(len), S2 = sum(len^2). Deterministic per-block partials.
// ---------------------------------------------------------------------------
__global__ __launch_bounds__(TPB)
void edge_kernel(const int* __restrict__ ei, const float2* __restrict__ pred,
                 int E, float* __restrict__ s1p, float* __restrict__ s2p) {
    const int gid    = blockIdx.x * TPB + threadIdx.x;
    const int stride = gridDim.x * TPB;
    float s1 = 0.0f, s2 = 0.0f;

    if ((E & 3) == 0) {
        for (int e = gid * 4; e < E; e += stride * 4) {
            int4 iv = *(const int4*)(ei + e);
            int4 jv = *(const int4*)(ei + E + e);
#define EDGE_ACC(II, JJ)                                                     \
            {                                                                \
                float2 pi = pred[(II)], pj = pred[(JJ)];                     \
                float dx = pi.x - pj.x, dy = pi.y - pj.y;                    \
                float l2 = dx * dx + dy * dy;                                \
                s2 += l2;                                                    \
                s1 += sqrtf(l2);                                             \
            }
            EDGE_ACC(iv.x, jv.x)
            EDGE_ACC(iv.y, jv.y)
            EDGE_ACC(iv.z, jv.z)
            EDGE_ACC(iv.w, jv.w)
#undef EDGE_ACC
        }
    } else {
        for (int e = gid; e < E; e += stride) {
            float2 pi = pred[ei[e]], pj = pred[ei[E + e]];
            float dx = pi.x - pj.x, dy = pi.y - pj.y;
            float l2 = dx * dx + dy * dy;
            s2 += l2;
            s1 += sqrtf(l2);
        }
    }

    __shared__ float sh1[NWAVES], sh2[NWAVES];
    float w1 = wave_reduce_wmma(s1);
    float w2 = wave_reduce_wmma(s2);
    const int lane = threadIdx.x & 31, wid = threadIdx.x >> 5;
    if (lane == 0) { sh1[wid] = w1; sh2[wid] = w2; }
    __syncthreads();
    if (threadIdx.x == 0) {
        float b1 = 0.0f, b2 = 0.0f;
        for (int w = 0; w < NWAVES; ++w) { b1 += sh1[w]; b2 += sh2[w]; }
        s1p[blockIdx.x] = b1;
        s2p[blockIdx.x] = b2;
    }
}

// ---------------------------------------------------------------------------
// Recon kernel: sum (pred - target)^2 over all 2N floats.
// ---------------------------------------------------------------------------
__global__ __launch_bounds__(TPB)
void recon_kernel(const float* __restrict__ pred, const float* __restrict__ targ,
                  int M, float* __restrict__ rp) {
    const int gid    = blockIdx.x * TPB + threadIdx.x;
    const int stride = gridDim.x * TPB;
    float acc = 0.0f;

    if ((M & 3) == 0) {
        for (int i = gid * 4; i < M; i += stride * 4) {
            float4 p = *(const float4*)(pred + i);
            float4 t = *(const float4*)(targ + i);
            float dx = p.x - t.x, dy = p.y - t.y, dz = p.z - t.z, dw = p.w - t.w;
            acc += dx * dx + dy * dy + dz * dz + dw * dw;
        }
    } else {
        for (int i = gid; i < M; i += stride) {
            float d = pred[i] - targ[i];
            acc += d * d;
        }
    }

    __shared__ float sh[NWAVES];
    float w = wave_reduce_wmma(acc);
    const int lane = threadIdx.x & 31, wid = threadIdx.x >> 5;
    if (lane == 0) sh[wid] = w;
    __syncthreads();
    if (threadIdx.x == 0) {
        float b = 0.0f;
        for (int q = 0; q < NWAVES; ++q) b += sh[q];
        rp[blockIdx.x] = b;
    }
}

// ---------------------------------------------------------------------------
// Drift kernel: batch_indices is sorted -> block g binary-searches its
// segment [start,end) and computes deterministic per-graph sums + count.
// ---------------------------------------------------------------------------
__global__ __launch_bounds__(TPB)
void drift_kernel(const float2* __restrict__ pred, const int* __restrict__ batch,
                  int N, float* __restrict__ gsum, float* __restrict__ gcnt) {
    const int g = blockIdx.x;

    int lo = 0, hi = N;
    while (lo < hi) { int mid = (lo + hi) >> 1; if (batch[mid] < g) lo = mid + 1; else hi = mid; }
    const int start = lo;
    hi = N;
    while (lo < hi) { int mid = (lo + hi) >> 1; if (batch[mid] < g + 1) lo = mid + 1; else hi = mid; }
    const int end = lo;

    float sx = 0.0f, sy = 0.0f;
    for (int n = start + threadIdx.x; n < end; n += TPB) {
        float2 p = pred[n];
        sx += p.x;
        sy += p.y;
    }

    __shared__ float shx[NWAVES], shy[NWAVES];
    float wx = wave_reduce_wmma(sx);
    float wy = wave_reduce_wmma(sy);
    const int lane = threadIdx.x & 31, wid = threadIdx.x >> 5;
    if (lane == 0) { shx[wid] = wx; shy[wid] = wy; }
    __syncthreads();
    if (threadIdx.x == 0) {
        float bx = 0.0f, by = 0.0f;
        for (int w = 0; w < NWAVES; ++w) { bx += shx[w]; by += shy[w]; }
        gsum[2 * g]     = bx;
        gsum[2 * g + 1] = by;
        gcnt[g]         = (float)(end - start);
    }
}

// ---------------------------------------------------------------------------
// KL kernel (tiny): sum over all G*L of (1 + lv - mu^2 - exp(lv)).
// ---------------------------------------------------------------------------
__global__ __launch_bounds__(TPB)
void kl_kernel(const float* __restrict__ mu, const float* __restrict__ lv,
               int M, float* __restrict__ klp) {
    float acc = 0.0f;
    for (int i = threadIdx.x; i < M; i += TPB) {
        float m = mu[i], l = lv[i];
        acc += 1.0f + l - m * m - expf(l);
    }
    __shared__ float sh[NWAVES];
    float w = wave_reduce_wmma(acc);
    const int lane = threadIdx.x & 31, wid = threadIdx.x >> 5;
    if (lane == 0) sh[wid] = w;
    __syncthreads();
    if (threadIdx.x == 0) {
        float b = 0.0f;
        for (int q = 0; q < NWAVES; ++q) b += sh[q];
        klp[0] = b;
    }
}

// ---------------------------------------------------------------------------
// Finalize: deterministic double-precision combine of all partials.
// ---------------------------------------------------------------------------
__global__ __launch_bounds__(TPB)
void finalize_kernel(const float* __restrict__ ws, const int* __restrict__ epoch,
                     float* __restrict__ out, int E, int N) {
    const float* s1p  = ws;
    const float* s2p  = ws + NBLK_E;
    const float* rp   = ws + 2 * NBLK_E;
    const float* gsum = ws + 2 * NBLK_E + NBLK_N;
    const float* gcnt = gsum + 2 * NGRAPH;
    const float* klp  = gcnt + NGRAPH;

    __shared__ double sd[4][TPB];
    const int t = threadIdx.x;

    double a1 = 0.0, a2 = 0.0, ar = 0.0, ad = 0.0;
    for (int i = t; i < NBLK_E; i += TPB) { a1 += (double)s1p[i]; a2 += (double)s2p[i]; }
    for (int i = t; i < NBLK_N; i += TPB) { ar += (double)rp[i]; }
    if (t < NGRAPH) {
        double c  = (double)gcnt[t];
        double mx = (double)gsum[2 * t] / c;
        double my = (double)gsum[2 * t + 1] / c;
        ad = mx * mx + my * my;
    }
    sd[0][t] = a1; sd[1][t] = a2; sd[2][t] = ar; sd[3][t] = ad;
    __syncthreads();
    for (int off = TPB / 2; off > 0; off >>= 1) {
        if (t < off)
            for (int q = 0; q < 4; ++q) sd[q][t] += sd[q][t + off];
        __syncthreads();
    }

    if (t == 0) {
        const double S1 = sd[0][0], S2 = sd[1][0], R = sd[2][0], D = sd[3][0];
        const double Ed = (double)E, Nd = (double)N;
        const double recon = R / (2.0 * Nd);
        const double lap   = S2 / Ed;
        const double arap  = (S2 - S1 * S1 / Ed) / (Ed - 1.0);
        const double drift = D / (double)NGRAPH;
        const double kl    = (-0.5 * (double)klp[0]) / (double)NGRAPH;
        const int ep = epoch[0];
        const double beta = (ep < 10) ? (double)ep * 0.1 : 1.0;
        out[0] = (float)(recon + 0.1 * lap + 0.01 * drift + 0.1 * arap + beta * kl);
    }
}

extern "C" void kernel_launch(void* const* d_in, const int* in_sizes, int n_in,
                              void* d_out, int out_size, void* d_ws, size_t ws_size,
                              hipStream_t stream) {
    const float* pred  = (const float*)d_in[0];
    const float* targ  = (const float*)d_in[1];
    const int*   ei    = (const int*)d_in[2];
    const int*   batch = (const int*)d_in[3];
    const float* mu    = (const float*)d_in[4];
    const float* lv    = (const float*)d_in[5];
    const int*   epoch = (const int*)d_in[6];
    float*       out   = (float*)d_out;
    float*       ws    = (float*)d_ws;

    const int N  = in_sizes[0] / 2;   // nodes
    const int E  = in_sizes[2] / 2;   // edges
    const int ML = in_sizes[4];       // NUM_GRAPHS * LATENT

    float* s1p  = ws;
    float* s2p  = ws + NBLK_E;
    float* rp   = ws + 2 * NBLK_E;
    float* gsum = ws + 2 * NBLK_E + NBLK_N;
    float* gcnt = gsum + 2 * NGRAPH;
    float* klp  = gcnt + NGRAPH;

    edge_kernel <<<NBLK_E, TPB, 0, stream>>>(ei, (const float2*)pred, E, s1p, s2p);
    recon_kernel<<<NBLK_N, TPB, 0, stream>>>(pred, targ, 2 * N, rp);
    drift_kernel<<<NGRAPH, TPB, 0, stream>>>((const float2*)pred, batch, N, gsum, gcnt);
    kl_kernel   <<<1,      TPB, 0, stream>>>(mu, lv, ML, klp);
    finalize_kernel<<<1,   TPB, 0, stream>>>(ws, epoch, out, E, N);
}